// E85InputAsMatrixCell_81819126989231
// MI455X (gfx1250) — compile-verified
//
#include <hip/hip_runtime.h>

typedef float v2f __attribute__((ext_vector_type(2)));
typedef float v8f __attribute__((ext_vector_type(8)));
typedef int   v4i_t __attribute__((vector_size(4 * sizeof(int))));

#define T_STEPS 512
#define BATCH   32
#define NS      64
#define DIM     4096
#define A_STRIDE  68  // 272B rows: 16B-aligned b128 fills, conflict-free b64 A-operand reads
#define MT_STRIDE 68  // transposed M: Mt[col][row]; conflict-free b64 B-operand reads

#if defined(__AMDGCN__) && __has_builtin(__builtin_amdgcn_global_load_async_to_lds_b128)
#define USE_ASYNC_LDS 1
typedef __attribute__((address_space(1))) v4i_t* gv4i_p;   // global int4*
typedef __attribute__((address_space(3))) v4i_t* lv4i_p;   // LDS int4*
#else
#define USE_ASYNC_LDS 0
#endif

__device__ __forceinline__ void wait_asynccnt0() {
#if __has_builtin(__builtin_amdgcn_s_wait_asynccnt)
    __builtin_amdgcn_s_wait_asynccnt(0);
#else
    asm volatile("s_wait_asynccnt 0x0" ::: "memory");
#endif
}

__global__ __launch_bounds__(256, 1) void e85_scan_kernel(
    const float* __restrict__ x,      // [T, B, DIM]
    const float* __restrict__ M0,     // [B, 64, 64]
    const float* __restrict__ scale_p,// [1]
    const float* __restrict__ gamma,  // [DIM]
    const float* __restrict__ beta,   // [DIM]
    float* __restrict__ out)          // [T*B*DIM] outputs ++ [B*DIM] M_final
{
    __shared__ float Abuf[2][NS * A_STRIDE];   // double-buffered A tile (row-major)
    __shared__ float Mt[NS * MT_STRIDE];       // resident state M, TRANSPOSED: Mt[col*68 + row]
    __shared__ float red[16];                  // block reduction scratch

    const int b    = blockIdx.x;
    const int tid  = threadIdx.x;
    const int wid  = tid >> 5;        // wave id 0..7 (wave32)
    const int lane = tid & 31;
    const int hh   = lane >> 4;       // half-wave 0/1
    const int l16  = lane & 15;
    const int mi   = wid & 3;         // C tile row index 0..3
    const int ni0  = (wid >> 2) * 2;  // first of two C tile cols {0,2}

    const float scale = scale_p[0];

    // ---- preload M0 transposed into LDS ----
    const float* M0b = M0 + (size_t)b * DIM;
    #pragma unroll
    for (int it = 0; it < 16; ++it) {
        int idx = it * 256 + tid;                 // row-major element index
        int row = idx >> 6, col = idx & 63;
        Mt[col * MT_STRIDE + row] = M0b[idx];
    }
    // ---- preload A[0] into buffer 0 (float4, fully coalesced) ----
    {
        const float4* xb = reinterpret_cast<const float4*>(x + (size_t)b * DIM);
        #pragma unroll
        for (int it = 0; it < 4; ++it) {
            int idx = it * 256 + tid;             // float4 index in [0,1024)
            float4 v = xb[idx];
            int row = idx >> 4, c4 = idx & 15;
            *reinterpret_cast<float4*>(&Abuf[0][row * A_STRIDE + c4 * 4]) = v;
        }
    }
    // ---- loop-invariant gamma/beta at this thread's C-layout positions ----
    const int rowbase = mi * 16 + 8 * hh;          // rows rowbase..rowbase+7
    const int col0 = ni0 * 16 + l16;
    const int col1 = col0 + 16;
    float g0[8], g1[8], bt0[8], bt1[8];
    #pragma unroll
    for (int v = 0; v < 8; ++v) {
        g0[v]  = gamma[(rowbase + v) * 64 + col0];
        g1[v]  = gamma[(rowbase + v) * 64 + col1];
        bt0[v] = beta [(rowbase + v) * 64 + col0];
        bt1[v] = beta [(rowbase + v) * 64 + col1];
    }
    __syncthreads();

#if !USE_ASYNC_LDS
    float4 pf[4];
#endif
    float nm0[8], nm1[8];

    for (int t = 0; t < T_STEPS; ++t) {
        const int cur = t & 1;
        const int nxt = cur ^ 1;

        // ---- prefetch x[t+1] into the back A buffer ----
        if (t + 1 < T_STEPS) {
            const float* xn = x + ((size_t)(t + 1) * BATCH + b) * DIM;
#if USE_ASYNC_LDS
            // CDNA5 async copy engine: global -> LDS with no VGPR staging,
            // tracked by ASYNCcnt; overlaps with the WMMA loop below.
            // (Safe: previous step's S3 ordered the last reads of Abuf[nxt].)
            #pragma unroll
            for (int it = 0; it < 4; ++it) {
                int idx = it * 256 + tid;          // float4 slot in [0,1024)
                int row = idx >> 4, c4 = idx & 15;
                __builtin_amdgcn_global_load_async_to_lds_b128(
                    (gv4i_p)(uintptr_t)(xn + idx * 4),
                    (lv4i_p)(uintptr_t)(&Abuf[nxt][row * A_STRIDE + c4 * 4]),
                    0, 0);
            }
#else
            const float4* xn4 = reinterpret_cast<const float4*>(xn);
            #pragma unroll
            for (int it = 0; it < 4; ++it) pf[it] = xn4[it * 256 + tid];
#endif
        }

        // ---- AM = A * M via V_WMMA_F32_16X16X4_F32, two 16x16 C tiles/wave ----
        v8f c0 = {0.f,0.f,0.f,0.f,0.f,0.f,0.f,0.f};
        v8f c1 = {0.f,0.f,0.f,0.f,0.f,0.f,0.f,0.f};
        const float* Ar  = &Abuf[cur][(mi * 16 + l16) * A_STRIDE];
        const float* Mc0 = &Mt[col0 * MT_STRIDE];
        const float* Mc1 = &Mt[col1 * MT_STRIDE];
        #pragma unroll
        for (int kk = 0; kk < 16; ++kk) {
            const int kb = kk * 4 + 2 * hh;        // K rows kb, kb+1 for this half
            v2f a  = *reinterpret_cast<const v2f*>(&Ar[kb]);   // A[m][kb..kb+1]
            v2f b0 = *reinterpret_cast<const v2f*>(&Mc0[kb]);  // M[kb..kb+1][col0]
            v2f b1 = *reinterpret_cast<const v2f*>(&Mc1[kb]);  // M[kb..kb+1][col1]
            c0 = __builtin_amdgcn_wmma_f32_16x16x4_f32(false, a, false, b0,
                                                       (short)0, c0, false, false);
            c1 = __builtin_amdgcn_wmma_f32_16x16x4_f32(false, a, false, b1,
                                                       (short)0, c1, false, false);
        }

        // ---- read old M at C positions (from transposed Mt) ----
        float mo0[8], mo1[8];
        #pragma unroll
        for (int v = 0; v < 8; ++v) {
            mo0[v] = Mc0[rowbase + v];
            mo1[v] = Mc1[rowbase + v];
        }

#if !USE_ASYNC_LDS
        // drain prefetched A[t+1] into the other buffer
        if (t + 1 < T_STEPS) {
            #pragma unroll
            for (int it = 0; it < 4; ++it) {
                int idx = it * 256 + tid;
                int row = idx >> 4, c4 = idx & 15;
                *reinterpret_cast<float4*>(&Abuf[nxt][row * A_STRIDE + c4 * 4]) = pf[it];
            }
        }
#endif

        // ---- M_new = M + scale*AM ; accumulate (sum, sumsq) ----
        float s0 = 0.f, s1 = 0.f;
        #pragma unroll
        for (int v = 0; v < 8; ++v) {
            float u0 = fmaf(scale, c0[v], mo0[v]);
            float u1 = fmaf(scale, c1[v], mo1[v]);
            nm0[v] = u0; nm1[v] = u1;
            s0 += u0 + u1;
            s1 = fmaf(u0, u0, s1);
            s1 = fmaf(u1, u1, s1);
        }
        // wave32 butterfly reduce
        #pragma unroll
        for (int off = 16; off > 0; off >>= 1) {
            s0 += __shfl_xor(s0, off, 32);
            s1 += __shfl_xor(s1, off, 32);
        }
        if (lane == 0) { red[wid * 2] = s0; red[wid * 2 + 1] = s1; }
        __syncthreads();   // S2: all Mt reads done; red partials visible
        float Ssum = 0.f, Ssq = 0.f;
        #pragma unroll
        for (int w = 0; w < 8; ++w) { Ssum += red[w * 2]; Ssq += red[w * 2 + 1]; }

        // Frobenius normalize + fused LayerNorm stats (biased var, like torch)
        const float r    = 1.0f / (sqrtf(Ssq) + 1e-6f);
        const float mu   = Ssum * r * (1.0f / 4096.0f);
        const float msq  = Ssq * r * r * (1.0f / 4096.0f);
        const float rstd = rsqrtf(msq - mu * mu + 1e-5f);

        float* outb = out + ((size_t)t * BATCH + b) * DIM;
        #pragma unroll
        for (int v = 0; v < 8; ++v) {
            float m0v = nm0[v] * r, m1v = nm1[v] * r;
            Mt[col0 * MT_STRIDE + rowbase + v] = m0v;
            Mt[col1 * MT_STRIDE + rowbase + v] = m1v;
            outb[(rowbase + v) * 64 + col0] = fmaf((m0v - mu) * rstd, g0[v], bt0[v]);
            outb[(rowbase + v) * 64 + col1] = fmaf((m1v - mu) * rstd, g1[v], bt1[v]);
            nm0[v] = m0v; nm1[v] = m1v;
        }

#if USE_ASYNC_LDS
        wait_asynccnt0();  // my async fills of Abuf[nxt] have landed in LDS
#endif
        __syncthreads();   // S3: Mt / Abuf[nxt] writes visible before next step reads
    }

    // ---- final M (normalized registers from last step) ----
    float* mf = out + (size_t)T_STEPS * BATCH * DIM + (size_t)b * DIM;
    #pragma unroll
    for (int v = 0; v < 8; ++v) {
        mf[(rowbase + v) * 64 + col0] = nm0[v];
        mf[(rowbase + v) * 64 + col1] = nm1[v];
    }
}

extern "C" void kernel_launch(void* const* d_in, const int* in_sizes, int n_in,
                              void* d_out, int out_size, void* d_ws, size_t ws_size,
                              hipStream_t stream) {
    (void)in_sizes; (void)n_in; (void)out_size; (void)d_ws; (void)ws_size;
    const float* x     = (const float*)d_in[0];
    const float* M0    = (const float*)d_in[1];
    const float* scale = (const float*)d_in[2];
    const float* gam   = (const float*)d_in[3];
    const float* bet   = (const float*)d_in[4];
    float* out = (float*)d_out;
    e85_scan_kernel<<<BATCH, 256, 0, stream>>>(x, M0, scale, gam, bet, out);
}